// GraphAttentionLayer_40407052321096
// MI455X (gfx1250) — compile-verified
//
#include <hip/hip_runtime.h>
#include <hip/hip_bf16.h>

// GAT layer for gfx1250 (MI455X).
// Sizes fixed by the reference:
#define NNODES 50000
#define NEDGES 800000
#define IN_F   128
#define OUT_F  32
#define HEADS  4          // H*O = 128 output features
#define NEG_SLOPE 0.2f
#define EPS 1e-10f

typedef __attribute__((ext_vector_type(2))) float v2f;
typedef __attribute__((ext_vector_type(8))) float v8f;

// ---------------------------------------------------------------------------
// Kernel 0: zero-fill
// ---------------------------------------------------------------------------
__global__ void zero_f32(float* __restrict__ p, int n) {
    int t = blockIdx.x * blockDim.x + threadIdx.x;
    if (t < n) p[t] = 0.0f;
}

// ---------------------------------------------------------------------------
// Kernel 1: projection GEMM  h[n][c] = sum_i x[n][i] * W[c/32][i][c%32]
// via V_WMMA_F32_16X16X4_F32.  One wave computes a 16-row x 128-col slab.
// Block = 256 threads (8 waves) -> 128 rows per block.
// W (64KB) is staged in LDS pre-swizzled into B-fragment order:
//   Wlds[kk][c] = (Wf[2kk][c], Wf[2kk+1][c])  so a B fragment is one b64 load.
// ---------------------------------------------------------------------------
__global__ __launch_bounds__(256) void gat_proj_wmma(
    const float* __restrict__ x, const float* __restrict__ W,
    float* __restrict__ h)
{
    __shared__ float2 Wlds[64 * 128];   // 64 KB
    const int tid = threadIdx.x;

    // Cooperative staged load of W, rearranged: Wf[k][c] = W[c>>5][k][c&31]
    for (int idx = tid; idx < 64 * 128; idx += 256) {
        int kk = idx >> 7;          // pair of K rows: 2kk, 2kk+1
        int c  = idx & 127;         // output column (head*32 + o)
        int base = (c >> 5) * (IN_F * OUT_F) + (2 * kk) * OUT_F + (c & 31);
        Wlds[idx] = make_float2(W[base], W[base + OUT_F]);
    }
    __syncthreads();

    const int wave = tid >> 5;
    const int lane = tid & 31;
    const int m    = lane & 15;     // A-row / B-col within tile
    const int kh   = lane >> 4;     // K half: 0 -> K={0,1}, 1 -> K={2,3}
    const int r0   = blockIdx.x * 128 + wave * 16;
    if (r0 >= NNODES) return;       // wave-uniform exit (N % 16 == 0)

    // Load full 16x128 A tile into registers: 32 K-fragments of 2 floats/lane.
    // Lane layout (ISA 7.12.2): lanes 0-15 hold K=4k+{0,1}, lanes 16-31 K=4k+{2,3}.
    float2 af[32];
    const float* xp = x + (r0 + m) * IN_F + kh * 2;
#pragma unroll
    for (int k = 0; k < 32; ++k)
        af[k] = *(const float2*)(xp + k * 4);

#pragma unroll 1
    for (int ct = 0; ct < 8; ++ct) {          // 8 column tiles of 16
        v8f acc = {};
#pragma unroll
        for (int k = 0; k < 32; ++k) {
            // B fragment: lanes 0-15 hold B[K=4k+0..1][N=m], lanes 16-31 K=4k+2..3
            float2 b2 = Wlds[(k * 2 + kh) * 128 + ct * 16 + m];
            v2f a; a.x = af[k].x; a.y = af[k].y;
            v2f b; b.x = b2.x;    b.y = b2.y;
            acc = __builtin_amdgcn_wmma_f32_16x16x4_f32(
                      false, a, false, b, (short)0, acc, false, false);
        }
        // C/D layout: VGPR v -> row v (lanes 0-15) / row v+8 (lanes 16-31), N = m
        const int col = ct * 16 + m;
        const int rbase = r0 + kh * 8;
#pragma unroll
        for (int v = 0; v < 8; ++v)
            h[(rbase + v) * 128 + col] = acc[v];
    }
}

// ---------------------------------------------------------------------------
// Kernel 2: node-level attention logits e_src/e_dst [N, H]
// ---------------------------------------------------------------------------
__global__ __launch_bounds__(256) void gat_logits(
    const float* __restrict__ h,
    const float* __restrict__ a_src, const float* __restrict__ a_dst,
    float* __restrict__ es, float* __restrict__ ed)
{
    int t = blockIdx.x * blockDim.x + threadIdx.x;
    if (t >= NNODES * HEADS) return;
    int n = t >> 2, hd = t & 3;
    const float* hp = h + n * 128 + hd * 32;
    const float* as = a_src + hd * 32;
    const float* ad = a_dst + hd * 32;
    float ss = 0.f, sd = 0.f;
#pragma unroll
    for (int o = 0; o < 32; ++o) {
        float v = hp[o];
        ss += v * as[o];
        sd += v * ad[o];
    }
    es[t] = ss;
    ed[t] = sd;
}

// ---------------------------------------------------------------------------
// Kernel 3: edge score + leaky-relu + segment max (clamped at 0).
// maxbuf starts at 0.0f; only v>0 can win, and non-negative floats order the
// same as their int bit patterns, so a signed-int atomicMax is exact.
// ---------------------------------------------------------------------------
__global__ __launch_bounds__(256) void gat_edge_max(
    const int* __restrict__ src, const int* __restrict__ dst,
    const float* __restrict__ es, const float* __restrict__ ed,
    float* __restrict__ ebuf, float* __restrict__ maxbuf)
{
    int t = blockIdx.x * blockDim.x + threadIdx.x;
    if (t >= NEDGES * HEADS) return;
    int e = t >> 2, hd = t & 3;
    int s = src[e], d = dst[e];
    float v = es[s * 4 + hd] + ed[d * 4 + hd];
    v = (v >= 0.f) ? v : NEG_SLOPE * v;
    ebuf[t] = v;
    if (v > 0.f)
        atomicMax((int*)&maxbuf[d * 4 + hd], __float_as_int(v));
}

// ---------------------------------------------------------------------------
// Kernel 4: exp(e - max[dst]) + segment sum
// ---------------------------------------------------------------------------
__global__ __launch_bounds__(256) void gat_exp_sum(
    const int* __restrict__ dst, const float* __restrict__ maxbuf,
    float* __restrict__ ebuf, float* __restrict__ sumbuf)
{
    int t = blockIdx.x * blockDim.x + threadIdx.x;
    if (t >= NEDGES * HEADS) return;
    int e = t >> 2, hd = t & 3;
    int d = dst[e];
    float ex = expf(ebuf[t] - maxbuf[d * 4 + hd]);
    ebuf[t] = ex;
    atomicAdd(&sumbuf[d * 4 + hd], ex);
}

// ---------------------------------------------------------------------------
// Kernel 5: normalize -> alpha (in place in ebuf)
// ---------------------------------------------------------------------------
__global__ __launch_bounds__(256) void gat_alpha(
    const int* __restrict__ dst, const float* __restrict__ sumbuf,
    float* __restrict__ ebuf)
{
    int t = blockIdx.x * blockDim.x + threadIdx.x;
    if (t >= NEDGES * HEADS) return;
    int e = t >> 2, hd = t & 3;
    int d = dst[e];
    ebuf[t] = ebuf[t] / (sumbuf[d * 4 + hd] + EPS);
}

// ---------------------------------------------------------------------------
// Kernel 6: aggregation. One wave per edge; lane l covers cols l+32j (head j).
// h and out are both L2-resident (25.6 MB each vs 192 MB L2), so the gathers
// and the global_atomic_add_f32 stream stay on-chip.
// ---------------------------------------------------------------------------
__global__ __launch_bounds__(256) void gat_aggregate(
    const int* __restrict__ src, const int* __restrict__ dst,
    const float* __restrict__ alpha, const float* __restrict__ h,
    float* __restrict__ out)
{
    int gw = (int)((blockIdx.x * blockDim.x + threadIdx.x) >> 5); // edge id
    int lane = threadIdx.x & 31;
    if (gw >= NEDGES) return;
    int s = src[gw], d = dst[gw];
    const float4 al = *(const float4*)(alpha + gw * 4);  // broadcast load
    const float* hs = h + s * 128;
    float* op = out + d * 128;
    atomicAdd(&op[lane],      al.x * hs[lane]);
    atomicAdd(&op[lane + 32], al.y * hs[lane + 32]);
    atomicAdd(&op[lane + 64], al.z * hs[lane + 64]);
    atomicAdd(&op[lane + 96], al.w * hs[lane + 96]);
}

// ---------------------------------------------------------------------------
// Launch. Workspace layout (floats), total ~41.6 MB:
//   h[6.4M] | es[200K] | ed[200K] | maxbuf[200K] | sumbuf[200K] | ebuf[3.2M]
// ---------------------------------------------------------------------------
extern "C" void kernel_launch(void* const* d_in, const int* in_sizes, int n_in,
                              void* d_out, int out_size, void* d_ws, size_t ws_size,
                              hipStream_t stream) {
    const float* x     = (const float*)d_in[0];
    const int*   edge  = (const int*)d_in[1];
    const float* W     = (const float*)d_in[2];
    const float* a_src = (const float*)d_in[3];
    const float* a_dst = (const float*)d_in[4];
    float* out = (float*)d_out;

    const int* src = edge;            // edge_index[0]
    const int* dst = edge + NEDGES;   // edge_index[1]

    float* h      = (float*)d_ws;
    float* es     = h + (size_t)NNODES * 128;
    float* ed     = es + (size_t)NNODES * HEADS;
    float* maxbuf = ed + (size_t)NNODES * HEADS;
    float* sumbuf = maxbuf + (size_t)NNODES * HEADS;
    float* ebuf   = sumbuf + (size_t)NNODES * HEADS;

    // zero out + (maxbuf,sumbuf) which are contiguous
    zero_f32<<<(NNODES * 128 + 255) / 256, 256, 0, stream>>>(out, NNODES * 128);
    zero_f32<<<(NNODES * 8 + 255) / 256, 256, 0, stream>>>(maxbuf, NNODES * 8);

    gat_proj_wmma<<<(NNODES + 127) / 128, 256, 0, stream>>>(x, W, h);
    gat_logits<<<(NNODES * HEADS + 255) / 256, 256, 0, stream>>>(h, a_src, a_dst, es, ed);
    gat_edge_max<<<(NEDGES * HEADS + 255) / 256, 256, 0, stream>>>(src, dst, es, ed, ebuf, maxbuf);
    gat_exp_sum<<<(NEDGES * HEADS + 255) / 256, 256, 0, stream>>>(dst, maxbuf, ebuf, sumbuf);
    gat_alpha<<<(NEDGES * HEADS + 255) / 256, 256, 0, stream>>>(dst, sumbuf, ebuf);
    gat_aggregate<<<(NEDGES * 32 + 255) / 256, 256, 0, stream>>>(src, dst, ebuf, h, out);
}